// YoloScaleLoss_11819749999131
// MI455X (gfx1250) — compile-verified
//
#include <hip/hip_runtime.h>
#include <math.h>

#define NB 16
#define NA 3
#define NG 64
#define NT 50
#define NCLS 91
#define NCH 96
#define CELLS (NB*NA*NG*NG)                // 196608
#define BLOCKS_PER_BATCH ((NA*NG*NG)/256)  // 48

typedef float v2f __attribute__((ext_vector_type(2)));
typedef float v8f __attribute__((ext_vector_type(8)));

// ANCHORS / STRIDE (STRIDE = 8 for SCALE=2); ANCH_MASK = {0,1,2} so
// m_anch == c_grid[0..2].
__constant__ float c_grid[9][2] = {
  {1.25f, 1.625f}, {2.0f, 3.75f},  {4.125f, 2.875f},
  {3.75f, 7.625f}, {7.75f, 5.625f},{7.375f, 14.875f},
  {14.5f, 11.25f}, {19.5f, 24.75f},{46.625f, 40.75f}};

struct LabelInfo {            // 64 bytes
  float tx, ty, tw, th;
  float valid, area;
  int   maskt, cell;
  float fx, fy, twt, tht;
  float tscale;
  int   cls, bestn, pad;
};

__device__ __forceinline__ float sp_(float x) {   // stable softplus
  return fmaxf(x, 0.f) + log1pf(expf(-fabsf(x)));
}
__device__ __forceinline__ float sig_(float x) { return 1.f / (1.f + expf(-x)); }

// ---------------------------------------------------------------- init
__global__ void k_init(int* map, float* loss) {
  int idx = blockIdx.x * blockDim.x + threadIdx.x;
  if (idx < CELLS) map[idx] = 0;
  if (idx == 0) loss[0] = 0.f;
}

// ---------------------------------------------------------------- labels
__global__ void k_labels(const float* __restrict__ labels, LabelInfo* __restrict__ L) {
  int l = blockIdx.x * blockDim.x + threadIdx.x;
  if (l >= NB * NT) return;
  const float* lb = labels + (size_t)l * 5;
  float cls = lb[0], x = lb[1], y = lb[2], w = lb[3], h = lb[4];
  float tx = x * NG, ty = y * NG, tw = w * NG, th = h * NG;
  float valid = (cls + x + y + w + h) > 0.f ? 1.f : 0.f;

  // anchor IoU argmax over 9 anchors (first max wins, like jnp.argmax)
  int bestk = 0; float best = -1.f;
  float areat = tw * th;
  for (int k = 0; k < 9; ++k) {
    float aw = c_grid[k][0], ah = c_grid[k][1];
    float mw = fminf(tw, aw), mh = fminf(th, ah);
    float inter = (mw > 0.f && mh > 0.f) ? mw * mh : 0.f;
    float iou = inter / (areat + aw * ah - inter);
    if (iou > best) { best = iou; bestk = k; }
  }
  int bestn = bestk % 3;
  int in_scale = (bestk < 3) ? 1 : 0;          // ANCH_MASK = {0,1,2}
  int maskt = (valid > 0.f) && in_scale;

  int i = (int)tx, j = (int)ty;                // floor (tx,ty > 0)
  int b = l / NT;
  int cell = ((b * NA + bestn) * NG + j) * NG + i;

  LabelInfo li;
  li.tx = tx; li.ty = ty; li.tw = tw; li.th = th;
  li.valid = valid; li.area = areat;
  li.maskt = maskt; li.cell = cell;
  li.fx = tx - floorf(tx); li.fy = ty - floorf(ty);
  li.twt = logf(tw / c_grid[bestn][0] + 1e-16f);
  li.tht = logf(th / c_grid[bestn][1] + 1e-16f);
  li.tscale = sqrtf(2.f - areat / (float)(NG * NG));
  li.cls = (int)cls; li.bestn = bestn; li.pad = 0;
  L[l] = li;
}

// ---------------------------------------------------------------- winner scatter
// Serialized per batch, in t order -> last write wins (matches scatter order).
__global__ void k_winner(const LabelInfo* __restrict__ L, int* __restrict__ map) {
  int b = threadIdx.x;
  if (b >= NB) return;
  for (int t = 0; t < NT; ++t) {
    int l = b * NT + t;
    if (L[l].maskt) map[L[l].cell] = l + 1;
  }
}

// ---------------------------------------------------------------- dense obj pass
__global__ void __launch_bounds__(256)
k_obj(const float* __restrict__ out, const LabelInfo* __restrict__ L,
      float* __restrict__ loss) {
  // Stage this batch's 50-label table straight into LDS with the CDNA5
  // async global->LDS path (no VGPR round-trip): 200 lanes x B128 = 3200 B.
  __shared__ __align__(16) float shL[NT * 16];
  int b = blockIdx.x / BLOCKS_PER_BATCH;
  int tid = threadIdx.x;
  if (tid < NT * 4) {
    unsigned lds = (unsigned)(size_t)(void*)&shL[0] + (unsigned)(tid * 16);
    unsigned long long ga =
        (unsigned long long)(size_t)((const char*)(L + b * NT) + tid * 16);
    asm volatile("global_load_async_to_lds_b128 %0, %1, off"
                 :: "v"(lds), "v"(ga) : "memory");
    asm volatile("s_wait_asynccnt 0" ::: "memory");
  }
  __syncthreads();
  const LabelInfo* Ls = (const LabelInfo*)shL;

  int cell = blockIdx.x * 256 + tid;
  int i = cell & (NG - 1);
  int j = (cell >> 6) & (NG - 1);
  int a = (cell >> 12) % NA;

  const float* row = out + (size_t)cell * NCH;
  float4 v = *reinterpret_cast<const float4*>(row);   // ch 0..3, 16B aligned
  float x4 = row[4];

  float px = sig_(v.x) + (float)i;
  float py = sig_(v.y) + (float)j;
  float pw = expf(v.z) * c_grid[a][0];
  float ph = expf(v.w) * c_grid[a][1];
  float hw = 0.5f * pw, hh = 0.5f * ph;
  float areap = pw * ph;

  float best = 0.f;
  for (int t = 0; t < NT; ++t) {
    float ttx = Ls[t].tx, tty = Ls[t].ty;
    float ttw = Ls[t].tw, tth = Ls[t].th;
    float tvv = Ls[t].valid;
    float tlx = fmaxf(px - hw, ttx - 0.5f * ttw);
    float tly = fmaxf(py - hh, tty - 0.5f * tth);
    float brx = fminf(px + hw, ttx + 0.5f * ttw);
    float bry = fminf(py + hh, tty + 0.5f * tth);
    float iw = brx - tlx, ih = bry - tly;
    float inter = (iw > 0.f && ih > 0.f) ? iw * ih : 0.f;
    float iou = inter / (areap + ttw * tth - inter);
    iou = (tvv != 0.f) ? iou : 0.f;
    best = fmaxf(best, iou);
  }
  float contrib = (best > 0.5f) ? 0.f : sp_(x4);      // obj_mask * softplus

  // ---- wave reduction via V_WMMA_F32_16X16X4_F32 (A = partials, B = ones)
  // A layout (32-bit, 16x4): lanes 0-15 -> (M=lane,K=0/1), lanes 16-31 -> K=2/3.
  // With A = {contrib, 0}, row m of D = contrib[m] + contrib[m+16].
  v2f av; av.x = contrib; av.y = 0.f;
  v2f bv; bv.x = 1.f;     bv.y = 1.f;                 // all-ones B (4x16)
  v8f c8 = {};
  v8f d = __builtin_amdgcn_wmma_f32_16x16x4_f32(
      false, av, false, bv, (short)0, c8, false, false);
  // lane 0-15 holds rowsums M=0..7 in d[0..7]; lane 16-31 holds M=8..15
  float half = d[0] + d[1] + d[2] + d[3] + d[4] + d[5] + d[6] + d[7];
  float tot = half + __shfl_xor(half, 16, 32);        // full 32-lane sum
  if ((tid & 31) == 0) atomicAdd(loss, tot);
}

// ---------------------------------------------------------------- sparse target pass
__global__ void k_targets(const float* __restrict__ out,
                          const LabelInfo* __restrict__ L,
                          const int* __restrict__ map,
                          float* __restrict__ loss) {
  int l = blockIdx.x * blockDim.x + threadIdx.x;
  if (l >= NB * NT) return;
  LabelInfo li = L[l];
  if (!li.maskt) return;
  if (map[li.cell] != l + 1) return;                  // only the winning label

  int b = l / NT;
  int cell = li.cell;
  int i = cell & (NG - 1);
  int j = (cell >> 6) & (NG - 1);
  int a = (cell >> 12) % NA;
  const float* row = out + (size_t)cell * NCH;
  __builtin_prefetch(row + 5, 0, 1);                  // cover the 91-ch gather

  float x0 = row[0], x1 = row[1], x2 = row[2], x3 = row[3], x4 = row[4];

  // loss_xy
  float lxy = (sp_(x0) - x0 * li.fx) + (sp_(x1) - x1 * li.fy);
  // loss_wh (out_wh - targ_wh)^2/2 = ts^2 * (x - t)^2 / 2
  float ts2 = li.tscale * li.tscale;
  float dw = x2 - li.twt, dh = x3 - li.tht;
  float lwh = 0.5f * ts2 * (dw * dw + dh * dh);
  // loss_cls: one-hot target
  float lcls = 0.f;
  for (int c = 0; c < NCLS; ++c) {
    float xc = row[5 + c];
    lcls += sp_(xc) - ((c == li.cls) ? xc : 0.f);
  }

  // obj correction: replace main-pass term with (softplus - x4) [obj_mask=1, t=1]
  float px = sig_(x0) + (float)i;
  float py = sig_(x1) + (float)j;
  float pw = expf(x2) * c_grid[a][0];
  float ph = expf(x3) * c_grid[a][1];
  float hw = 0.5f * pw, hh = 0.5f * ph;
  float areap = pw * ph;
  float best = 0.f;
  for (int t = 0; t < NT; ++t) {
    LabelInfo lt = L[b * NT + t];
    float tlx = fmaxf(px - hw, lt.tx - 0.5f * lt.tw);
    float tly = fmaxf(py - hh, lt.ty - 0.5f * lt.th);
    float brx = fminf(px + hw, lt.tx + 0.5f * lt.tw);
    float bry = fminf(py + hh, lt.ty + 0.5f * lt.th);
    float iw = brx - tlx, ih = bry - tly;
    float inter = (iw > 0.f && ih > 0.f) ? iw * ih : 0.f;
    float iou = inter / (areap + lt.tw * lt.th - inter);
    iou = (lt.valid != 0.f) ? iou : 0.f;
    best = fmaxf(best, iou);
  }
  float mainTerm = (best > 0.5f) ? 0.f : sp_(x4);
  float corr = (sp_(x4) - x4) - mainTerm;

  atomicAdd(loss, lxy + lwh + lcls + corr);
}

// ---------------------------------------------------------------- launch
extern "C" void kernel_launch(void* const* d_in, const int* in_sizes, int n_in,
                              void* d_out, int out_size, void* d_ws, size_t ws_size,
                              hipStream_t stream) {
  const float* output = (const float*)d_in[0];   // (16,3,64,64,96) f32
  const float* labels = (const float*)d_in[1];   // (16,50,5) f32
  float* loss = (float*)d_out;                   // scalar

  int* map = (int*)d_ws;                                         // CELLS ints
  LabelInfo* L = (LabelInfo*)((char*)d_ws + (size_t)CELLS * 4);  // 800 * 64B

  k_init   <<<CELLS / 256, 256, 0, stream>>>(map, loss);
  k_labels <<<(NB * NT + 255) / 256, 256, 0, stream>>>(labels, L);
  k_winner <<<1, 32, 0, stream>>>(L, map);
  k_obj    <<<CELLS / 256, 256, 0, stream>>>(output, L, loss);
  k_targets<<<(NB * NT + 255) / 256, 256, 0, stream>>>(output, L, map, loss);
}